// S4Layer_35253091565780
// MI455X (gfx1250) — compile-verified
//
#include <hip/hip_runtime.h>
#include <hip/hip_bf16.h>

#define B_   16
#define H_   128
#define L_   4096
#define N_   64
#define M2L  8192            // 2*L, FFT length for the convolution
#define PI_F 3.14159265358979f

typedef __attribute__((ext_vector_type(2))) float v2f;
typedef __attribute__((ext_vector_type(8))) float v8f;
typedef int async_v4i __attribute__((vector_size(16)));   // builtin's pointee type

#if __has_builtin(__builtin_amdgcn_global_load_async_to_lds_b128)
#define HAVE_ASYNC_LDS 1
#endif

// ---------------------------------------------------------------------------
// Kernel 0: twiddle table tw[j] = exp(-2*pi*i*j/8192), j in [0,4096)
// ---------------------------------------------------------------------------
__global__ __launch_bounds__(256) void twiddle_kernel(float2* __restrict__ tw) {
    int i = blockIdx.x * blockDim.x + threadIdx.x;
    if (i < M2L / 2) {
        float ang = -2.0f * PI_F * (float)i / (float)M2L;
        float s, c;
        sincosf(ang, &s, &c);
        tw[i] = make_float2(c, s);
    }
}

// ---------------------------------------------------------------------------
// Kernel 1: Cauchy stage with V_WMMA_F32_16X16X4_F32.
// Per head h: Ks[l,k] = sum_n den[h,l,n] * V[h,n,k]  realified as
// [L, 2N] x [2N, 8] GEMM; one wave owns a 16-row l tile and accumulates 32
// WMMAs (K=4 each) over 2N=128. Head-invariant operands (clipped Lambda and
// the realified V-matrix in exact lane layout) are staged in LDS once per
// block so the inner loop is ds_load + rcp + wmma only.
// ---------------------------------------------------------------------------
__global__ __launch_bounds__(256) void cauchy_kernel(
    const float* __restrict__ Lre, const float* __restrict__ Lim,
    const float* __restrict__ Pri, const float* __restrict__ Bri,
    const float* __restrict__ Cri, const float* __restrict__ logstep,
    float2* __restrict__ atRoots)
{
    __shared__ float lamS[N_][2];        // (-min(Lre,-1e-4), Lim)     0.5 KB
    __shared__ float bvS[32][32][2];     // [chunk][lane][re/im]         8 KB
    __shared__ float red[8][16][8];      // per-wave spill of D cols 0..7 4 KB

    const int h    = blockIdx.y;
    const int tid  = threadIdx.x;
    const int wave = tid >> 5;
    const int lane = tid & 31;
    const int m    = lane & 15;          // row within 16-l tile / B column
    const int half = lane >> 4;          // selects n = 2c / 2c+1
    const int l0   = (blockIdx.x * 8 + wave) * 16;
    const int l    = l0 + m;

    // ---- per-block staging of head-invariant operands ----
    if (tid < N_) {
        lamS[tid][0] = -fminf(Lre[h * N_ + tid], -1e-4f);  // = dre > 0
        lamS[tid][1] = Lim[h * N_ + tid];
    }
    for (int idx = tid; idx < 32 * 32; idx += 256) {
        const int c  = idx >> 5;
        const int ln = idx & 31;
        const int hf = ln >> 4;
        const int j  = ln & 15;          // B column (only 0..7 live)
        float bx = 0.0f, by = 0.0f;
        if (j < 8) {
            const int n    = 2 * c + hf;
            const int base = (h * N_ + n) * 2;
            const float pr = Pri[base], pi = Pri[base + 1];
            const float br = Bri[base], bi = Bri[base + 1];
            const float cr = Cri[base], ci = Cri[base + 1];
            const int k = j >> 1;        // product: 0=C*B 1=C*P 2=P*B 3=P*P
            // conj(x)*y: re = xr*yr + xi*yi ; im = xr*yi - xi*yr
            const float xr = (k < 2) ? cr : pr;
            const float xi = (k < 2) ? ci : pi;
            const float yr = (k & 1) ? pr : br;
            const float yi = (k & 1) ? pi : bi;
            const float vr = xr * yr + xi * yi;
            const float vi = xr * yi - xi * yr;
            const bool isim = (j & 1);
            bx = isim ? vi : vr;         // dr-row entry
            by = isim ? vr : -vi;        // di-row entry
        }
        bvS[c][ln][0] = bx;
        bvS[c][ln][1] = by;
    }
    __syncthreads();

    const float stp = __expf(logstep[h]);
    // g = i*t, t = (2/step)*tan(pi*l/L)   (purely imaginary after clipping)
    float sn, cs;
    __sincosf(PI_F * (float)l * (1.0f / (float)L_), &sn, &cs);
    const float t = 2.0f * __builtin_amdgcn_rcpf(stp) * sn *
                    __builtin_amdgcn_rcpf(cs);

    v8f acc = {};                        // C/D accumulator (16x16 f32)

#pragma unroll 8
    for (int c = 0; c < 32; ++c) {
        const int n = 2 * c + half;
        // A operand: den = 1/(g - Lambda_n) for this lane's (l, n)
        const float dre = lamS[n][0];
        const float dim = t - lamS[n][1];
        const float inv = __builtin_amdgcn_rcpf(dre * dre + dim * dim);
        v2f a;
        a.x = dre * inv;                 // Re(den)
        a.y = -dim * inv;                // Im(den)
        v2f bv;
        bv.x = bvS[c][lane][0];
        bv.y = bvS[c][lane][1];
        acc = __builtin_amdgcn_wmma_f32_16x16x4_f32(
            false, a, false, bv, (short)0, acc, false, false);
    }

    // Spill the 8 live columns: VGPR r holds row (r + 8*half), col = lane&15.
    if (m < 8) {
#pragma unroll
        for (int r = 0; r < 8; ++r)
            red[wave][r + 8 * half][m] = acc[r];
    }
    __syncthreads();

    if (lane < 16) {
        const float k00r = red[wave][lane][0], k00i = red[wave][lane][1];
        const float k01r = red[wave][lane][2], k01i = red[wave][lane][3];
        const float k10r = red[wave][lane][4], k10i = red[wave][lane][5];
        const float k11r = red[wave][lane][6], k11i = red[wave][lane][7];

        const float wr = 1.0f + k11r, wi = k11i;
        const float winv = __builtin_amdgcn_rcpf(wr * wr + wi * wi);
        const float prd = k01r * k10r - k01i * k10i;
        const float pim = k01r * k10i + k01i * k10r;
        const float qr = (prd * wr + pim * wi) * winv;
        const float qi = (pim * wr - prd * wi) * winv;
        const float rr = k00r - qr, ri = k00i - qi;

        const int ll = l0 + lane;
        float s2, c2;
        __sincosf(PI_F * (float)ll * (1.0f / (float)L_), &s2, &c2);
        const float tc = s2 * __builtin_amdgcn_rcpf(c2);   // c = 1 + i*tc
        atRoots[(size_t)h * L_ + ll] =
            make_float2(rr - tc * ri, ri + tc * rr);
    }
}

// ---------------------------------------------------------------------------
// In-LDS iterative radix-2 DIT FFT (forward). Twiddles indexed from an
// 8192-point table of 4096 entries: w(pos,len) = tw[pos * (8192/len)]  (len <= 8192).
// ---------------------------------------------------------------------------
__device__ __forceinline__ void fft_shared(float2* s, const float2* tw,
                                           int M, int logM, int tid, int nthr)
{
    // bit-reversal permutation (pairs are disjoint; no sync needed inside)
    for (int i = tid; i < M; i += nthr) {
        int j = (int)(__brev((unsigned)i) >> (32 - logM));
        if (i < j) { float2 tmp = s[i]; s[i] = s[j]; s[j] = tmp; }
    }
    __syncthreads();

    const int nbf = M >> 1;
    for (int sh = 1; sh <= logM; ++sh) {
        const int half = 1 << (sh - 1);
        const int step = M2L >> sh;          // twiddle stride in the table
        for (int k = tid; k < nbf; k += nthr) {
            const int pos = k & (half - 1);
            const int grp = k >> (sh - 1);
            const int i0  = (grp << sh) + pos;
            const int i1  = i0 + half;
            const float2 w = tw[pos * step];
            const float2 b = s[i1];
            const float  tr = b.x * w.x - b.y * w.y;
            const float  ti = b.x * w.y + b.y * w.x;
            const float2 a = s[i0];
            s[i0] = make_float2(a.x + tr, a.y + ti);
            s[i1] = make_float2(a.x - tr, a.y - ti);
        }
        __syncthreads();
    }
}

// ---------------------------------------------------------------------------
// Kernel 2: per head. K = Re(ifft_L(atRoots)); Kd = fft_2L(pad(K)).
// ifft via conj trick: ifft(x) = conj(fft(conj(x)))/L.
// ---------------------------------------------------------------------------
__global__ __launch_bounds__(256) void kfft_kernel(
    const float2* __restrict__ ar, const float2* __restrict__ twg,
    float2* __restrict__ kd)
{
    extern __shared__ char smem[];
    float2* s   = (float2*)smem;                          // 8192 entries
    float2* twl = (float2*)(smem + M2L * sizeof(float2)); // 4096 entries

    const int tid = threadIdx.x, nthr = blockDim.x;
    const int h = blockIdx.x;

    for (int i = tid; i < M2L / 2; i += nthr) twl[i] = twg[i];
    for (int i = tid; i < L_; i += nthr) {
        float2 v = ar[(size_t)h * L_ + i];
        s[i] = make_float2(v.x, -v.y);                  // conj for inverse
    }
    __syncthreads();

    fft_shared(s, twl, L_, 12, tid, nthr);              // length-4096 FFT

    const float sc = 1.0f / (float)L_;
    for (int i = tid; i < M2L; i += nthr) {             // K, zero-padded to 2L
        const float kr = (i < L_) ? s[i].x * sc : 0.0f; // Re(conj(.))/L
        s[i] = make_float2(kr, 0.0f);
    }
    __syncthreads();

    fft_shared(s, twl, M2L, 13, tid, nthr);             // length-8192 FFT

    for (int i = tid; i < M2L; i += nthr)
        kd[(size_t)h * M2L + i] = s[i];
}

// ---------------------------------------------------------------------------
// Kernel 3: per (b,h) sequence. y = Re(ifft(fft(pad(u)) .* Kd))[:L] + D*u.
// Kd is pulled into LDS with GLOBAL_LOAD_ASYNC_TO_LDS_B128 (ASYNCcnt),
// fully hidden behind the first FFT.
// ---------------------------------------------------------------------------
__global__ __launch_bounds__(256) void conv_kernel(
    const float* __restrict__ u, const float2* __restrict__ kd,
    const float2* __restrict__ twg, const float* __restrict__ Dp,
    float* __restrict__ y)
{
    extern __shared__ char smem[];
    float2* s    = (float2*)smem;                                  // 8192
    float2* twl  = (float2*)(smem + M2L * sizeof(float2));         // 4096
    float2* kbuf = (float2*)(smem + (M2L + M2L / 2) * sizeof(float2)); // 8192

    const int tid = threadIdx.x, nthr = blockDim.x;
    const int bh = blockIdx.x;
    const int h  = bh % H_;
    const float* up = u + (size_t)bh * L_;
    const float2* kh = kd + (size_t)h * M2L;

#if defined(HAVE_ASYNC_LDS)
    // Kick off the 64KB Kd copy into LDS; it overlaps the first FFT.
    {
        __attribute__((address_space(1))) async_v4i* gk =
            (__attribute__((address_space(1))) async_v4i*)kh;
        __attribute__((address_space(3))) async_v4i* lk =
            (__attribute__((address_space(3))) async_v4i*)kbuf;
#pragma unroll
        for (int k = 0; k < 16; ++k) {
            const int i = tid + k * 256;                // 16B granules
            __builtin_amdgcn_global_load_async_to_lds_b128(gk + i, lk + i,
                                                           0, 0);
        }
    }
#else
    __builtin_prefetch(kh, 0, 0);                       // global_prefetch_b8
#endif

    float ureg[16];
    for (int i = tid; i < M2L / 2; i += nthr) twl[i] = twg[i];
#pragma unroll
    for (int k = 0; k < 16; ++k) {
        const int i = tid + k * 256;
        const float v = up[i];
        ureg[k] = v;
        s[i]      = make_float2(v, 0.0f);
        s[i + L_] = make_float2(0.0f, 0.0f);
    }
    __syncthreads();

    fft_shared(s, twl, M2L, 13, tid, nthr);             // forward FFT of u

#if defined(HAVE_ASYNC_LDS)
#if __has_builtin(__builtin_amdgcn_s_wait_asynccnt)
    __builtin_amdgcn_s_wait_asynccnt(0);
#else
    asm volatile("s_wait_asynccnt 0x0" ::: "memory");
#endif
    __syncthreads();
#endif

    for (int i = tid; i < M2L; i += nthr) {             // pointwise * Kd, conj
        const float2 a = s[i];
#if defined(HAVE_ASYNC_LDS)
        const float2 b = kbuf[i];
#else
        const float2 b = kh[i];
#endif
        const float pr = a.x * b.x - a.y * b.y;
        const float pi = a.x * b.y + a.y * b.x;
        s[i] = make_float2(pr, -pi);                    // conj -> inverse FFT
    }
    __syncthreads();

    fft_shared(s, twl, M2L, 13, tid, nthr);             // "inverse" FFT

    const float dh = Dp[h];
    const float sc = 1.0f / (float)M2L;
#pragma unroll
    for (int k = 0; k < 16; ++k) {
        const int i = tid + k * 256;
        y[(size_t)bh * L_ + i] = s[i].x * sc + dh * ureg[k];
    }
}

// ---------------------------------------------------------------------------
// Host-side launch. Inputs (setup_inputs order):
// 0:u 1:Lambda_re 2:Lambda_im 3:P_ri 4:B_ri 5:C 6:log_step 7:D
// ---------------------------------------------------------------------------
extern "C" void kernel_launch(void* const* d_in, const int* in_sizes, int n_in,
                              void* d_out, int out_size, void* d_ws, size_t ws_size,
                              hipStream_t stream) {
    const float* u     = (const float*)d_in[0];
    const float* Lre   = (const float*)d_in[1];
    const float* Lim   = (const float*)d_in[2];
    const float* Pri   = (const float*)d_in[3];
    const float* Bri   = (const float*)d_in[4];
    const float* Cri   = (const float*)d_in[5];
    const float* lstep = (const float*)d_in[6];
    const float* Dp    = (const float*)d_in[7];
    float* y = (float*)d_out;

    char* ws = (char*)d_ws;
    float2* tw = (float2*)ws;                                   // 32 KB
    float2* ar = (float2*)(ws + 32768);                         // 4 MB
    float2* kd = (float2*)(ws + 32768 + (size_t)H_ * L_ * 8);   // 8 MB

    twiddle_kernel<<<(M2L / 2 + 255) / 256, 256, 0, stream>>>(tw);

    dim3 g1(L_ / 128, H_);
    cauchy_kernel<<<g1, 256, 0, stream>>>(Lre, Lim, Pri, Bri, Cri, lstep, ar);

    const size_t shmem_k = (size_t)(M2L + M2L / 2) * sizeof(float2);  // 96 KB
    const size_t shmem_c = (size_t)(2 * M2L + M2L / 2) * sizeof(float2); // 160 KB
    kfft_kernel<<<H_, 256, shmem_k, stream>>>(ar, tw, kd);
    conv_kernel<<<B_ * H_, 256, shmem_c, stream>>>(u, kd, tw, Dp, y);
}